// TestModel_35510789603586
// MI455X (gfx1250) — compile-verified
//
#include <hip/hip_runtime.h>
#include <math.h>

typedef float v2f __attribute__((ext_vector_type(2)));
typedef float v8f __attribute__((ext_vector_type(8)));

#define N_NODES  50000
#define N_EDGES  800000
#define N_GRAPHS 256
#define HEADS    2
#define NEG_SLOPE 0.2f
#define EPS_F 1e-16f
#define GEMM_WAVES 5            // 3125 M-tiles = 5 * 625 -> no remainder

// ---------------------------------------------------------------- utilities
__global__ void fill_kernel(float* __restrict__ p, float v, int n) {
  int i = blockIdx.x * blockDim.x + threadIdx.x;
  int stride = gridDim.x * blockDim.x;
  for (; i < n; i += stride) p[i] = v;
}

__device__ __forceinline__ void atomicMaxF(float* addr, float v) {
  // IEEE-754 monotonic bit trick: works for mixed signs, init must be -inf.
  if (v >= 0.f) atomicMax((int*)addr, __float_as_int(v));
  else          atomicMin((unsigned int*)addr, __float_as_uint(v));
}

// ------------------------------------------------- fp32 WMMA tile GEMM
// C[M x Ncols] = A[M x K] * B_head[K x Ncols_head]; grid=(M/16/WAVES, N/16, H).
// Block = GEMM_WAVES wave32s sharing one LDS-staged B tile [K x 16]
// (pair-interleaved so each fragment read is one ds_load_b64).
// Layouts per CDNA5 ISA 7.12.2:
//   A 16x4 f32 : lanes 0-15 -> K={k,k+1}, lanes 16-31 -> K={k+2,k+3}, M = lane&15
//   B 4x16 f32 : mirrored, N = lane&15
//   C/D 16x16  : VGPR j -> row j (lanes 0-15) / row j+8 (lanes 16-31), N = lane&15
__global__ void wmma_gemm_f32(const float* __restrict__ A,
                              const float* __restrict__ B,
                              const float* __restrict__ bias,
                              float* __restrict__ C,
                              int M, int K, int lda, int ldb, int ldc,
                              int bHeadStride, int cHeadOff, int applyElu) {
  __shared__ float ldsB[64 * 16];          // K <= 64 in this model (4 KB)
  const int tid  = threadIdx.x;
  const int wave = tid >> 5;
  const int lane = tid & 31;
  const int half = lane >> 4;              // 0: K-low pair, 1: K-high pair
  const int l    = lane & 15;
  const int col0 = blockIdx.y * 16;
  const float* Bh = B + (size_t)blockIdx.z * bHeadStride;

  // cooperative stage of B tile, pair-interleaved: [(k>>1)*32 + col*2 + (k&1)]
  for (int t = tid; t < K * 16; t += blockDim.x) {
    const int k = t >> 4, c = t & 15;
    ldsB[(k >> 1) * 32 + c * 2 + (k & 1)] = Bh[(size_t)k * ldb + col0 + c];
  }
  __syncthreads();

  const int row0 = (blockIdx.x * GEMM_WAVES + wave) * 16;
  const float* arow = A + (size_t)(row0 + l) * lda;
  const int colB = col0 + l;
  const int colC = blockIdx.z * cHeadOff + colB;

  v8f acc = {0.f, 0.f, 0.f, 0.f, 0.f, 0.f, 0.f, 0.f};
  for (int k = 0; k < K; k += 4) {
    const int kk = k + half * 2;           // even
    v2f av, bv;
    av.x = arow[kk];
    av.y = arow[kk + 1];
    const float* bp = &ldsB[(kk >> 1) * 32 + l * 2];
    bv.x = bp[0];
    bv.y = bp[1];
    // 8-arg form: (neg_a, A, neg_b, B, c_mod, C, reuse_a, reuse_b)
    acc = __builtin_amdgcn_wmma_f32_16x16x4_f32(false, av, false, bv,
                                                (short)0, acc, false, false);
  }
#pragma unroll
  for (int j = 0; j < 8; ++j) {
    const int r = row0 + j + half * 8;
    float v = acc[j];
    if (bias) v += bias[colB];
    if (applyElu) v = v > 0.f ? v : (expf(v) - 1.f);
    if (r < M) C[(size_t)r * ldc + colC] = v;
  }
}

// --------------------------------------------- per-(node,head) attention dots
__global__ void node_scores_kernel(const float* __restrict__ ht,    // [N,H,Cout]
                                   const float* __restrict__ a_src, // [H,Cout]
                                   const float* __restrict__ a_dst,
                                   float* __restrict__ esrc, float* __restrict__ edst,
                                   int Cout) {
  int idx = blockIdx.x * blockDim.x + threadIdx.x;  // n*H + h
  if (idx >= N_NODES * HEADS) return;
  const int h = idx & (HEADS - 1);
  const float* hv = ht + (size_t)idx * Cout;
  const float* as = a_src + h * Cout;
  const float* ad = a_dst + h * Cout;
  float s = 0.f, d = 0.f;
  for (int c = 0; c < Cout; c += 4) {
    const float4 x  = *(const float4*)&hv[c];
    const float4 a1 = *(const float4*)&as[c];
    const float4 a2 = *(const float4*)&ad[c];
    s += x.x * a1.x + x.y * a1.y + x.z * a1.z + x.w * a1.w;
    d += x.x * a2.x + x.y * a2.y + x.z * a2.z + x.w * a2.w;
  }
  esrc[idx] = s; edst[idx] = d;
}

// -------------------------------------------------- edge pass 1: leakyrelu+max
__global__ void edge_e_kernel(const int* __restrict__ src, const int* __restrict__ dst,
                              const float* __restrict__ esrc, const float* __restrict__ edst,
                              float* __restrict__ ebuf, float* __restrict__ emax) {
  int idx = blockIdx.x * blockDim.x + threadIdx.x;  // e*H + h
  if (idx >= N_EDGES * HEADS) return;
  const int e = idx >> 1;           // HEADS == 2
  const int h = idx & 1;
  const int s = src[e], d = dst[e];
  float v = esrc[s * HEADS + h] + edst[d * HEADS + h];
  v = v > 0.f ? v : NEG_SLOPE * v;
  ebuf[idx] = v;
  atomicMaxF(&emax[d * HEADS + h], v);
}

// -------------------------------------------------- edge pass 2: exp + denom
__global__ void edge_w_kernel(const int* __restrict__ dst,
                              float* __restrict__ ebuf,
                              const float* __restrict__ emax,
                              float* __restrict__ denom) {
  int idx = blockIdx.x * blockDim.x + threadIdx.x;
  if (idx >= N_EDGES * HEADS) return;
  const int e = idx >> 1;
  const int h = idx & 1;
  const int d = dst[e];
  float m = emax[d * HEADS + h];
  if (!isfinite(m)) m = 0.f;
  const float w = expf(ebuf[idx] - m);
  ebuf[idx] = w;
  atomicAdd(&denom[d * HEADS + h], w);
}

// ---------------------- edge pass 3: weighted message scatter (4 ch / thread)
__global__ void edge_agg_kernel(const int* __restrict__ src, const int* __restrict__ dst,
                                const float* __restrict__ ht,     // [N,H,Cout]
                                const float* __restrict__ wbuf,   // [E,H]
                                const float* __restrict__ denom,  // [N,H]
                                float* __restrict__ agg,          // [N,H,Cout]
                                int Cout) {
  const int cq = Cout >> 2;
  const long total = (long)N_EDGES * HEADS * cq;
  long idx = (long)blockIdx.x * blockDim.x + threadIdx.x;
  if (idx >= total) return;
  const int  q  = (int)(idx % cq);
  const long eh = idx / cq;
  const int  h  = (int)(eh & 1);
  const int  e  = (int)(eh >> 1);
  const int  s  = src[e], d = dst[e];
  const float alpha = wbuf[eh] / (denom[d * HEADS + h] + EPS_F);
  const float4 hv = *(const float4*)&ht[((size_t)s * HEADS + h) * Cout + 4 * q];
  float* ap = &agg[((size_t)d * HEADS + h) * Cout + 4 * q];
  atomicAdd(ap + 0, hv.x * alpha);
  atomicAdd(ap + 1, hv.y * alpha);
  atomicAdd(ap + 2, hv.z * alpha);
  atomicAdd(ap + 3, hv.w * alpha);
}

// ---------------------------------------- head mean + bias + ELU -> next input
__device__ __forceinline__ float eluf(float v) {
  return v > 0.f ? v : (expf(v) - 1.f);
}

__global__ void head_mean_elu_kernel(const float* __restrict__ agg,  // [N,H,Cout]
                                     const float* __restrict__ bias, // [Cout]
                                     float* __restrict__ out,        // [N,Cout]
                                     int Cout) {
  const int cq = Cout >> 2;
  long idx = (long)blockIdx.x * blockDim.x + threadIdx.x;  // n*cq + q
  if (idx >= (long)N_NODES * cq) return;
  const int  q = (int)(idx % cq);
  const long n = idx / cq;
  const float4 a0 = *(const float4*)&agg[((size_t)n * HEADS) * Cout + 4 * q];
  const float4 a1 = *(const float4*)&agg[((size_t)n * HEADS + 1) * Cout + 4 * q];
  const float4 b  = *(const float4*)&bias[4 * q];
  float4 r;
  r.x = eluf(0.5f * (a0.x + a1.x) + b.x);
  r.y = eluf(0.5f * (a0.y + a1.y) + b.y);
  r.z = eluf(0.5f * (a0.z + a1.z) + b.z);
  r.w = eluf(0.5f * (a0.w + a1.w) + b.w);
  *(float4*)&out[(size_t)n * Cout + 4 * q] = r;
}

// --------------------------------------------------------- global mean pool
__global__ void pool_kernel(const float* __restrict__ h2,    // [N,128]
                            const int* __restrict__ batch,
                            float* __restrict__ sums,        // [G,128]
                            float* __restrict__ cnt) {       // [G]
  long idx = (long)blockIdx.x * blockDim.x + threadIdx.x;    // n*32 + q
  if (idx >= (long)N_NODES * 32) return;
  const int q = (int)(idx % 32);
  const int n = (int)(idx / 32);
  const int g = batch[n];
  const float4 v = *(const float4*)&h2[(size_t)n * 128 + 4 * q];
  float* sp = &sums[g * 128 + 4 * q];
  atomicAdd(sp + 0, v.x);
  atomicAdd(sp + 1, v.y);
  atomicAdd(sp + 2, v.z);
  atomicAdd(sp + 3, v.w);
  if (q == 0) atomicAdd(&cnt[g], 1.0f);
}

// ---------------------------------------------------------- final projection
__global__ void final_kernel(const float* __restrict__ sums, const float* __restrict__ cnt,
                             const float* __restrict__ lo_w, const float* __restrict__ lo_b,
                             float* __restrict__ out) {
  int idx = blockIdx.x * blockDim.x + threadIdx.x;   // g*24 + o
  if (idx >= N_GRAPHS * 24) return;
  const int o = idx % 24;
  const int g = idx / 24;
  float cn = cnt[g]; cn = cn > 1.f ? cn : 1.f;
  const float invc = 1.f / cn;
  float acc = 0.f;
  for (int c = 0; c < 128; ++c) acc += sums[g * 128 + c] * invc * lo_w[c * 24 + o];
  out[idx] = acc + lo_b[o];
}

// ============================================================== launch driver
extern "C" void kernel_launch(void* const* d_in, const int* in_sizes, int n_in,
                              void* d_out, int out_size, void* d_ws, size_t ws_size,
                              hipStream_t stream) {
  (void)in_sizes; (void)n_in; (void)out_size; (void)ws_size;
  const float* x      = (const float*)d_in[0];
  const int*   eidx   = (const int*)d_in[1];
  const int*   batch  = (const int*)d_in[2];
  const float* fc0_w  = (const float*)d_in[3];
  const float* fc0_b  = (const float*)d_in[4];
  const float* W1     = (const float*)d_in[5];
  const float* a_src1 = (const float*)d_in[6];
  const float* a_dst1 = (const float*)d_in[7];
  const float* b1     = (const float*)d_in[8];
  const float* W2     = (const float*)d_in[9];
  const float* a_src2 = (const float*)d_in[10];
  const float* a_dst2 = (const float*)d_in[11];
  const float* b2     = (const float*)d_in[12];
  const float* lo_w   = (const float*)d_in[13];
  const float* lo_b   = (const float*)d_in[14];
  float* out = (float*)d_out;

  const int* src = eidx;
  const int* dst = eidx + N_EDGES;

  float* ws = (float*)d_ws;
  size_t off = 0;
  auto alloc = [&](size_t n) { float* p = ws + off; off += n; return p; };
  float* h0   = alloc((size_t)N_NODES * 32);            // fc0 output
  float* h1t  = alloc((size_t)N_NODES * HEADS * 64);    // layer1 transform
  float* esrc = alloc((size_t)N_NODES * HEADS);
  float* edst = alloc((size_t)N_NODES * HEADS);
  float* emax = alloc((size_t)N_NODES * HEADS);
  float* den  = alloc((size_t)N_NODES * HEADS);
  float* ebuf = alloc((size_t)N_EDGES * HEADS);
  float* agg1 = alloc((size_t)N_NODES * HEADS * 64);
  float* h1   = alloc((size_t)N_NODES * 64);
  float* h2t  = alloc((size_t)N_NODES * HEADS * 128);
  float* agg2 = alloc((size_t)N_NODES * HEADS * 128);
  float* h2   = alloc((size_t)N_NODES * 128);
  float* sums = alloc((size_t)N_GRAPHS * 128);
  float* cnt  = alloc((size_t)N_GRAPHS);

  const int TB = 256;
  const int GB = GEMM_WAVES * 32;             // 160 threads, 5 wave32s
  const int MT = (N_NODES / 16) / GEMM_WAVES; // 625 blocks along M
  const float NEG_INF = -__builtin_huge_valf();

  // ---- fc0: h0 = elu(x @ fc0_w + fc0_b), [50000,16]x[16,32]
  {
    dim3 g(MT, 32 / 16, 1);
    wmma_gemm_f32<<<g, GB, 0, stream>>>(x, fc0_w, fc0_b, h0,
                                        N_NODES, 16, 16, 32, 32, 0, 0, 1);
  }

  // ---- GAT layer 1 -------------------------------------------------------
  {
    dim3 g(MT, 64 / 16, HEADS);   // per-head [50000,32]x[32,64]
    wmma_gemm_f32<<<g, GB, 0, stream>>>(h0, W1, nullptr, h1t,
                                        N_NODES, 32, 32, 64, HEADS * 64,
                                        32 * 64, 64, 0);
  }
  node_scores_kernel<<<(N_NODES * HEADS + TB - 1) / TB, TB, 0, stream>>>(
      h1t, a_src1, a_dst1, esrc, edst, 64);
  fill_kernel<<<256, TB, 0, stream>>>(emax, NEG_INF, N_NODES * HEADS);
  fill_kernel<<<256, TB, 0, stream>>>(den, 0.f, N_NODES * HEADS);
  edge_e_kernel<<<(N_EDGES * HEADS + TB - 1) / TB, TB, 0, stream>>>(
      src, dst, esrc, edst, ebuf, emax);
  edge_w_kernel<<<(N_EDGES * HEADS + TB - 1) / TB, TB, 0, stream>>>(
      dst, ebuf, emax, den);
  fill_kernel<<<4096, TB, 0, stream>>>(agg1, 0.f, N_NODES * HEADS * 64);
  {
    long total = (long)N_EDGES * HEADS * (64 / 4);
    edge_agg_kernel<<<(unsigned)((total + TB - 1) / TB), TB, 0, stream>>>(
        src, dst, h1t, ebuf, den, agg1, 64);
  }
  {
    long total = (long)N_NODES * (64 / 4);
    head_mean_elu_kernel<<<(unsigned)((total + TB - 1) / TB), TB, 0, stream>>>(
        agg1, b1, h1, 64);
  }

  // ---- GAT layer 2 -------------------------------------------------------
  {
    dim3 g(MT, 128 / 16, HEADS);  // per-head [50000,64]x[64,128]
    wmma_gemm_f32<<<g, GB, 0, stream>>>(h1, W2, nullptr, h2t,
                                        N_NODES, 64, 64, 128, HEADS * 128,
                                        64 * 128, 128, 0);
  }
  node_scores_kernel<<<(N_NODES * HEADS + TB - 1) / TB, TB, 0, stream>>>(
      h2t, a_src2, a_dst2, esrc, edst, 128);
  fill_kernel<<<256, TB, 0, stream>>>(emax, NEG_INF, N_NODES * HEADS);
  fill_kernel<<<256, TB, 0, stream>>>(den, 0.f, N_NODES * HEADS);
  edge_e_kernel<<<(N_EDGES * HEADS + TB - 1) / TB, TB, 0, stream>>>(
      src, dst, esrc, edst, ebuf, emax);
  edge_w_kernel<<<(N_EDGES * HEADS + TB - 1) / TB, TB, 0, stream>>>(
      dst, ebuf, emax, den);
  fill_kernel<<<8192, TB, 0, stream>>>(agg2, 0.f, N_NODES * HEADS * 128);
  {
    long total = (long)N_EDGES * HEADS * (128 / 4);
    edge_agg_kernel<<<(unsigned)((total + TB - 1) / TB), TB, 0, stream>>>(
        src, dst, h2t, ebuf, den, agg2, 128);
  }
  {
    long total = (long)N_NODES * (128 / 4);
    head_mean_elu_kernel<<<(unsigned)((total + TB - 1) / TB), TB, 0, stream>>>(
        agg2, b2, h2, 128);
  }

  // ---- pool + output -----------------------------------------------------
  fill_kernel<<<64, TB, 0, stream>>>(sums, 0.f, N_GRAPHS * 128 + N_GRAPHS);
  {
    long total = (long)N_NODES * 32;
    pool_kernel<<<(unsigned)((total + TB - 1) / TB), TB, 0, stream>>>(
        h2, batch, sums, cnt);
  }
  final_kernel<<<(N_GRAPHS * 24 + TB - 1) / TB, TB, 0, stream>>>(
      sums, cnt, lo_w, lo_b, out);
}